// CombinedModel_11897059410621
// MI455X (gfx1250) — compile-verified
//
#include <hip/hip_runtime.h>

typedef float v2f __attribute__((ext_vector_type(2)));
typedef float v8f __attribute__((ext_vector_type(8)));

#define N_NODES  100000
#define N_EDGES  3200000
#define N_GRAPHS 2048
#define FIN      9
#define FIN_PAD  16
#define HDIM     64
#define EXTRA_F  32
#define BN_EPS   1e-5f

static inline unsigned cdiv(long long a, long long b) { return (unsigned)((a + b - 1) / b); }

// ---------------- degree / normalization ----------------
__global__ void init_deg_kernel(float* __restrict__ deg, int n) {
  int i = blockIdx.x * blockDim.x + threadIdx.x;
  if (i < n) deg[i] = 1.0f;  // self-loop contribution
}

__global__ void deg_accum_kernel(const int* __restrict__ dst, float* __restrict__ deg, int e) {
  int i = blockIdx.x * blockDim.x + threadIdx.x;
  if (i < e) atomicAdd(&deg[dst[i]], 1.0f);
}

__global__ void deg_rsqrt_kernel(float* __restrict__ deg, int n) {
  int i = blockIdx.x * blockDim.x + threadIdx.x;
  if (i < n) deg[i] = rsqrtf(deg[i]);  // deg >= 1 always (self-loops)
}

// ---------------- padding helpers (make K a multiple of 4, no K guards in GEMM) ----
__global__ void pad_x_kernel(const float* __restrict__ x, float* __restrict__ xp, int n) {
  int i = blockIdx.x * blockDim.x + threadIdx.x;
  if (i >= n * FIN_PAD) return;
  const int node = i >> 4, c = i & 15;
  xp[i] = (c < FIN) ? x[node * FIN + c] : 0.0f;
}

__global__ void pad_w1_kernel(const float* __restrict__ w, float* __restrict__ wp) {
  int i = blockIdx.x * blockDim.x + threadIdx.x;
  if (i >= FIN_PAD * HDIM) return;
  const int r = i >> 6, c = i & 63;
  wp[i] = (r < FIN) ? w[r * HDIM + c] : 0.0f;
}

// ---------------- f32 WMMA GEMM: C[M,Ncol] = A[M,K] @ B[K,Ncol] (+bias) ----------------
// block = 128 threads (4 waves). Block computes a 64x(NT*16) output tile: wave w owns
// rows [blkRow + w*16, +16) and all NT column subtiles (NT accumulators, one shared A
// fragment). B window staged in LDS as (even-k, odd-k) float2 pairs -> one ds_load_b64
// per B fragment. NT is compile-time so the k-loop is a branch-free wmma sequence.
// grid = (ceil(M/64), Ncol/(NT*16)). Requires K % 4 == 0, K <= 128, Ncol % (NT*16) == 0.
template <int NT>
__global__ void wmma_gemm_f32(const float* __restrict__ A, const float* __restrict__ B,
                              const float* __restrict__ bias, float* __restrict__ C,
                              int M, int K, int Ncol) {
  constexpr int W = NT * 16;     // column window width
  __shared__ v2f Bs[64 * W];     // [K/2][W] pairs; K <= 128 -> <= 32 KB

  const int tid  = threadIdx.x;
  const int lane = tid & 31;
  const int wave = tid >> 5;
  const int half = lane >> 4;    // 0: lanes 0-15, 1: lanes 16-31
  const int l    = lane & 15;
  const int colBase = blockIdx.y * W;

  // cooperative B stage: Bs[kp*W + c] = { B[2kp][colBase+c], B[2kp+1][colBase+c] }
  const int npairs = (K >> 1) * W;
  for (int idx = tid; idx < npairs; idx += 128) {
    const int kp = idx / W, c = idx % W;   // W is a power of two -> shifts
    const long long bc = (long long)colBase + c;
    v2f v;
    v.x = B[(long long)(2 * kp    ) * Ncol + bc];
    v.y = B[(long long)(2 * kp + 1) * Ncol + bc];
    Bs[idx] = v;
  }
  __syncthreads();

  const int rowTile = blockIdx.x * 64 + wave * 16;
  int ar = rowTile + l;          // A row for this lane (clamped; store is guarded)
  if (ar >= M) ar = M - 1;
  const float* __restrict__ Arow = A + (long long)ar * K;

  v8f acc[NT];
  #pragma unroll
  for (int ct = 0; ct < NT; ++ct) acc[ct] = (v8f){};

  for (int kk = 0; kk < K; kk += 4) {
    // A fragment: contiguous float2 at column kk + half*2  (single b64 load)
    const v2f a = *(const v2f*)(Arow + kk + half * 2);
    const int kp = (kk >> 1) + half;       // LDS pair row for this half-wave
    const v2f* __restrict__ brow = Bs + kp * W + l;
    #pragma unroll
    for (int ct = 0; ct < NT; ++ct) {
      const v2f b = brow[ct * 16];
      acc[ct] = __builtin_amdgcn_wmma_f32_16x16x4_f32(false, a, false, b, (short)0,
                                                      acc[ct], false, false);
    }
  }

  // store: VGPR r -> row rowTile + r + half*8, col colBase + ct*16 + l
  #pragma unroll
  for (int ct = 0; ct < NT; ++ct) {
    const int col = colBase + ct * 16 + l;
    const float bv = bias ? bias[col] : 0.0f;
    #pragma unroll
    for (int r = 0; r < 8; ++r) {
      const int row = rowTile + r + half * 8;
      if (row < M) C[(long long)row * Ncol + col] = acc[ct][r] + bv;
    }
  }
}

// ---------------- GCN aggregation ----------------
// Q[n] = dis[n]^2 * P[n]   (self-loop term; full overwrite initializes accumulator)
__global__ void selfloop_init_kernel(const float* __restrict__ xw, const float* __restrict__ dis,
                                     float* __restrict__ agg, int n) {
  int i = blockIdx.x * blockDim.x + threadIdx.x;          // n*16 threads, float4 chunks
  if (i >= n * 16) return;
  const int node = i >> 4, c = i & 15;
  const float s = dis[node] * dis[node];
  const float4 v = ((const float4*)xw)[(long long)node * 16 + c];
  float4 o; o.x = v.x * s; o.y = v.y * s; o.z = v.z * s; o.w = v.w * s;
  ((float4*)agg)[(long long)node * 16 + c] = o;
}

// agg[dst] += dis[src]*dis[dst] * xw[src]
__global__ void edge_agg_kernel(const float* __restrict__ xw, const float* __restrict__ dis,
                                const int* __restrict__ src, const int* __restrict__ dst,
                                float* __restrict__ agg, int e) {
  int i = blockIdx.x * blockDim.x + threadIdx.x;          // e*16 threads, float4 chunks
  if (i >= e * 16) return;
  const int ei = i >> 4, c = i & 15;
  const int s = src[ei], d = dst[ei];
  const float coef = dis[s] * dis[d];
  const float4 v = ((const float4*)xw)[(long long)s * 16 + c];
  float* out = agg + (long long)d * HDIM + c * 4;
  atomicAdd(out + 0, v.x * coef);
  atomicAdd(out + 1, v.y * coef);
  atomicAdd(out + 2, v.z * coef);
  atomicAdd(out + 3, v.w * coef);
}

__global__ void bias_relu_kernel(float* __restrict__ h, const float* __restrict__ b, int total) {
  int i = blockIdx.x * blockDim.x + threadIdx.x;
  if (i >= total) return;
  const float v = h[i] + b[i & (HDIM - 1)];
  h[i] = v > 0.0f ? v : 0.0f;
}

// ---------------- pooling / concat ----------------
__global__ void zero_f32_kernel(float* __restrict__ p, int n) {
  int i = blockIdx.x * blockDim.x + threadIdx.x;
  if (i < n) p[i] = 0.0f;
}

__global__ void pool_kernel(const float* __restrict__ h, const int* __restrict__ batch,
                            float* __restrict__ emb, float* __restrict__ cnt, int n) {
  int i = blockIdx.x * blockDim.x + threadIdx.x;          // n*16 threads
  if (i >= n * 16) return;
  const int node = i >> 4, c = i & 15;
  const int g = batch[node];
  const float4 v = ((const float4*)h)[(long long)node * 16 + c];
  float* out = emb + (long long)g * HDIM + c * 4;
  atomicAdd(out + 0, v.x); atomicAdd(out + 1, v.y);
  atomicAdd(out + 2, v.z); atomicAdd(out + 3, v.w);
  if (c == 0) atomicAdd(&cnt[g], 1.0f);
}

__global__ void concat_kernel(const float* __restrict__ emb, const float* __restrict__ cnt,
                              const float* __restrict__ extra, float* __restrict__ z) {
  int i = blockIdx.x * blockDim.x + threadIdx.x;          // G * (HDIM+EXTRA_F)
  const int W = HDIM + EXTRA_F;
  if (i >= N_GRAPHS * W) return;
  const int g = i / W, j = i - g * W;
  z[i] = (j < HDIM) ? emb[g * HDIM + j] / fmaxf(cnt[g], 1.0f)
                    : extra[g * EXTRA_F + (j - HDIM)];
}

// ---------------- BatchNorm (training-mode batch stats) ----------------
__global__ void bn_stats_kernel(const float* __restrict__ z, int rows, int F,
                                float* __restrict__ mu, float* __restrict__ istd) {
  __shared__ float ssum[256];
  __shared__ float ssq[256];
  const int f = blockIdx.x, t = threadIdx.x;
  float s = 0.0f, q = 0.0f;
  for (int r = t; r < rows; r += blockDim.x) {
    const float v = z[(long long)r * F + f];
    s += v; q += v * v;
  }
  ssum[t] = s; ssq[t] = q;
  __syncthreads();
  for (int o = 128; o > 0; o >>= 1) {
    if (t < o) { ssum[t] += ssum[t + o]; ssq[t] += ssq[t + o]; }
    __syncthreads();
  }
  if (t == 0) {
    const float m = ssum[0] / rows;
    const float var = ssq[0] / rows - m * m;
    mu[f] = m;
    istd[f] = rsqrtf(var + BN_EPS);
  }
}

__global__ void bn_apply_kernel(float* __restrict__ z, const float* __restrict__ mu,
                                const float* __restrict__ istd, const float* __restrict__ g,
                                const float* __restrict__ be, int total, int F) {
  int i = blockIdx.x * blockDim.x + threadIdx.x;
  if (i >= total) return;
  const int f = i % F;
  const float v = (z[i] - mu[f]) * istd[f] * g[f] + be[f];
  z[i] = v > 0.0f ? v : 0.0f;
}

__global__ void final_kernel(const float* __restrict__ z, const float* __restrict__ W,
                             const float* __restrict__ b, float* __restrict__ out, int rows) {
  int gIdx = blockIdx.x * blockDim.x + threadIdx.x;
  if (gIdx >= rows) return;
  float s = b[0];
  #pragma unroll
  for (int j = 0; j < 32; ++j) s += z[gIdx * 32 + j] * W[j];
  out[gIdx] = s;
}

// ---------------- launch ----------------
extern "C" void kernel_launch(void* const* d_in, const int* in_sizes, int n_in,
                              void* d_out, int out_size, void* d_ws, size_t ws_size,
                              hipStream_t stream) {
  const float* x     = (const float*)d_in[0];
  const int*   eidx  = (const int*)d_in[1];
  const int*   batch = (const int*)d_in[2];
  const float* extra = (const float*)d_in[3];
  const float* W1 = (const float*)d_in[4];  const float* b1 = (const float*)d_in[5];
  const float* W2 = (const float*)d_in[6];  const float* b2 = (const float*)d_in[7];
  const float* W3 = (const float*)d_in[8];  const float* b3 = (const float*)d_in[9];
  const float* Wm0 = (const float*)d_in[10]; const float* bm0 = (const float*)d_in[11];
  const float* g0  = (const float*)d_in[12]; const float* be0 = (const float*)d_in[13];
  const float* Wm1 = (const float*)d_in[14]; const float* bm1 = (const float*)d_in[15];
  const float* g1  = (const float*)d_in[16]; const float* be1 = (const float*)d_in[17];
  const float* Wm2 = (const float*)d_in[18]; const float* bm2 = (const float*)d_in[19];
  const float* g2  = (const float*)d_in[20]; const float* be2 = (const float*)d_in[21];
  const float* Wm3 = (const float*)d_in[22]; const float* bm3 = (const float*)d_in[23];
  float* out = (float*)d_out;

  const int* src = eidx;            // edge_index[0]
  const int* dst = eidx + N_EDGES;  // edge_index[1]

  // workspace carve-up (floats)
  float* ws  = (float*)d_ws;
  float* P   = ws;                                  // N*H  (xw)
  float* Q   = P + (size_t)N_NODES * HDIM;          // N*H  (agg / h)
  float* dis = Q + (size_t)N_NODES * HDIM;          // N
  float* emb = dis + N_NODES;                       // G*H
  float* cnt = emb + (size_t)N_GRAPHS * HDIM;       // G   (contiguous after emb)
  float* z0  = cnt + N_GRAPHS;                      // G*96
  float* z1  = z0 + (size_t)N_GRAPHS * (HDIM + EXTRA_F);  // G*128
  float* z2  = z1 + (size_t)N_GRAPHS * 128;         // G*64
  float* z3  = z2 + (size_t)N_GRAPHS * 64;          // G*32
  float* mu  = z3 + (size_t)N_GRAPHS * 32;          // 128
  float* istd = mu + 128;                           // 128
  float* Xp  = istd + 128;                          // N*16 (padded x)
  float* W1p = Xp + (size_t)N_NODES * FIN_PAD;      // 16*64 (padded W1)

  const int T = 256;

  // ---- GCN normalization coefficients ----
  init_deg_kernel<<<cdiv(N_NODES, T), T, 0, stream>>>(dis, N_NODES);
  deg_accum_kernel<<<cdiv(N_EDGES, T), T, 0, stream>>>(dst, dis, N_EDGES);
  deg_rsqrt_kernel<<<cdiv(N_NODES, T), T, 0, stream>>>(dis, N_NODES);

  // ---- pad x and W1 so GEMM needs no K guards ----
  pad_x_kernel<<<cdiv((long long)N_NODES * FIN_PAD, T), T, 0, stream>>>(x, Xp, N_NODES);
  pad_w1_kernel<<<cdiv(FIN_PAD * HDIM, T), T, 0, stream>>>(W1, W1p);

  const dim3 gemmBlk(128);
  const unsigned gRows = cdiv(N_NODES, 64);

  // ---- layer 1: Xp[N,16] @ W1p[16,64] ----
  wmma_gemm_f32<4><<<dim3(gRows, 1), gemmBlk, 0, stream>>>(Xp, W1p, nullptr, P, N_NODES, FIN_PAD, HDIM);
  selfloop_init_kernel<<<cdiv((long long)N_NODES * 16, T), T, 0, stream>>>(P, dis, Q, N_NODES);
  edge_agg_kernel<<<cdiv((long long)N_EDGES * 16, T), T, 0, stream>>>(P, dis, src, dst, Q, N_EDGES);
  bias_relu_kernel<<<cdiv((long long)N_NODES * HDIM, T), T, 0, stream>>>(Q, b1, N_NODES * HDIM);

  // ---- layer 2 ----
  wmma_gemm_f32<4><<<dim3(gRows, 1), gemmBlk, 0, stream>>>(Q, W2, nullptr, P, N_NODES, HDIM, HDIM);
  selfloop_init_kernel<<<cdiv((long long)N_NODES * 16, T), T, 0, stream>>>(P, dis, Q, N_NODES);
  edge_agg_kernel<<<cdiv((long long)N_EDGES * 16, T), T, 0, stream>>>(P, dis, src, dst, Q, N_EDGES);
  bias_relu_kernel<<<cdiv((long long)N_NODES * HDIM, T), T, 0, stream>>>(Q, b2, N_NODES * HDIM);

  // ---- layer 3 ----
  wmma_gemm_f32<4><<<dim3(gRows, 1), gemmBlk, 0, stream>>>(Q, W3, nullptr, P, N_NODES, HDIM, HDIM);
  selfloop_init_kernel<<<cdiv((long long)N_NODES * 16, T), T, 0, stream>>>(P, dis, Q, N_NODES);
  edge_agg_kernel<<<cdiv((long long)N_EDGES * 16, T), T, 0, stream>>>(P, dis, src, dst, Q, N_EDGES);
  bias_relu_kernel<<<cdiv((long long)N_NODES * HDIM, T), T, 0, stream>>>(Q, b3, N_NODES * HDIM);

  // ---- global mean pool + concat ----
  zero_f32_kernel<<<cdiv(N_GRAPHS * (HDIM + 1), T), T, 0, stream>>>(emb, N_GRAPHS * (HDIM + 1));
  pool_kernel<<<cdiv((long long)N_NODES * 16, T), T, 0, stream>>>(Q, batch, emb, cnt, N_NODES);
  concat_kernel<<<cdiv(N_GRAPHS * (HDIM + EXTRA_F), T), T, 0, stream>>>(emb, cnt, extra, z0);

  // ---- MLP with training-mode BatchNorm ----
  const unsigned gG = cdiv(N_GRAPHS, 64);
  // layer 0: [G,96] @ [96,128]
  wmma_gemm_f32<4><<<dim3(gG, 2), gemmBlk, 0, stream>>>(z0, Wm0, bm0, z1, N_GRAPHS, 96, 128);
  bn_stats_kernel<<<128, 256, 0, stream>>>(z1, N_GRAPHS, 128, mu, istd);
  bn_apply_kernel<<<cdiv(N_GRAPHS * 128, T), T, 0, stream>>>(z1, mu, istd, g0, be0, N_GRAPHS * 128, 128);
  // layer 1: [G,128] @ [128,64]
  wmma_gemm_f32<4><<<dim3(gG, 1), gemmBlk, 0, stream>>>(z1, Wm1, bm1, z2, N_GRAPHS, 128, 64);
  bn_stats_kernel<<<64, 256, 0, stream>>>(z2, N_GRAPHS, 64, mu, istd);
  bn_apply_kernel<<<cdiv(N_GRAPHS * 64, T), T, 0, stream>>>(z2, mu, istd, g1, be1, N_GRAPHS * 64, 64);
  // layer 2: [G,64] @ [64,32]  (NT=2 instance, 32-wide window)
  wmma_gemm_f32<2><<<dim3(gG, 1), gemmBlk, 0, stream>>>(z2, Wm2, bm2, z3, N_GRAPHS, 64, 32);
  bn_stats_kernel<<<32, 256, 0, stream>>>(z3, N_GRAPHS, 32, mu, istd);
  bn_apply_kernel<<<cdiv(N_GRAPHS * 32, T), T, 0, stream>>>(z3, mu, istd, g2, be2, N_GRAPHS * 32, 32);
  // head: [G,32] @ [32,1]
  final_kernel<<<cdiv(N_GRAPHS, T), T, 0, stream>>>(z3, Wm3, bm3, out, N_GRAPHS);
}